// HR2HK_9552007266522
// MI455X (gfx1250) — compile-verified
//
#include <hip/hip_runtime.h>

#define NORB   9
#define NATOMS 400
#define NEDGES 6400
#define NK     4
#define ADIM   (NATOMS * NORB)   // 3600
#define PAIRD  81
#define NODED  58

// Precomputed gather maps (DIMS = [1,3,5]); replicates _build_maps() exactly.
__constant__ int PAIR_IDX[81] = {
   0,  1,  2,  3,  4,  5,  6,  7,  8,
   9, 12, 13, 14, 21, 22, 23, 24, 25,
  10, 15, 16, 17, 26, 27, 28, 29, 30,
  11, 18, 19, 20, 31, 32, 33, 34, 35,
  36, 41, 42, 43, 56, 57, 58, 59, 60,
  37, 44, 45, 46, 61, 62, 63, 64, 65,
  38, 47, 48, 49, 66, 67, 68, 69, 70,
  39, 50, 51, 52, 71, 72, 73, 74, 75,
  40, 53, 54, 55, 76, 77, 78, 79, 80
};

__constant__ int NODE_IDX[81] = {
   0,  1,  2,  3,  4,  5,  6,  7,  8,
   1,  9, 10, 11, 18, 19, 20, 21, 22,
   2, 12, 13, 14, 23, 24, 25, 26, 27,
   3, 15, 16, 17, 28, 29, 30, 31, 32,
   4, 18, 23, 28, 33, 34, 35, 36, 37,
   5, 19, 24, 29, 38, 39, 40, 41, 42,
   6, 20, 25, 30, 43, 44, 45, 46, 47,
   7, 21, 26, 31, 48, 49, 50, 51, 52,
   8, 22, 27, 32, 53, 54, 55, 56, 57
};

typedef float v2f __attribute__((ext_vector_type(2)));
typedef float v8f __attribute__((ext_vector_type(8)));

// ---------------------------------------------------------------------------
// 1) Zero-fill H (415 MB stream; the dominant cost -> wide float4 stores).
// ---------------------------------------------------------------------------
__global__ void zero_kernel(float4* __restrict__ out, long n4) {
    long i      = (long)blockIdx.x * blockDim.x + threadIdx.x;
    long stride = (long)gridDim.x * blockDim.x;
    const float4 z = make_float4(0.f, 0.f, 0.f, 0.f);
    for (; i < n4; i += stride) out[i] = z;
}

// ---------------------------------------------------------------------------
// 2) phase[k][e] = exp(-2*pi*i * dot(kpoints[k], shifts[e])) via WMMA.
//    D(16x16) = A(16x4) x B(4x16):  A rows 0..3 = kpoints (col 3 zero-padded),
//    B cols = 16 edges of this tile (row 3 zero-padded). One wave per tile.
//    A f32 16x4 layout (ISA 7.12.2): lanes 0-15 M=lane, v0={K0|K2}, v1={K1|K3}.
//    D f32 16x16 layout: VGPR v, lanes 0-15 -> (M=v, N=lane).
// ---------------------------------------------------------------------------
__global__ __launch_bounds__(32) void phase_wmma_kernel(
    const float* __restrict__ kpoints,   // [NK][3]
    const float* __restrict__ shifts,    // [NEDGES][3]
    float2*      __restrict__ phase)     // [NK][NEDGES]
{
    const int tile = blockIdx.x;            // NEDGES/16 = 400 tiles
    const int lane = threadIdx.x;           // 0..31
    const int half = lane >> 4;             // 0: K in {0,1}; 1: K in {2,3}
    const int m    = lane & 15;

    // A-matrix: kpoints, zero-padded to 16x4
    float a0 = 0.f, a1 = 0.f;
    if (m < NK) {
        if (half == 0) { a0 = kpoints[m * 3 + 0]; a1 = kpoints[m * 3 + 1]; }
        else           { a0 = kpoints[m * 3 + 2]; a1 = 0.f;                }
    }
    v2f a; a.x = a0; a.y = a1;

    // B-matrix: B[d][n] = shifts[tile*16+n][d], row K=3 zero
    const int e = tile * 16 + m;
    float b0, b1;
    if (half == 0) { b0 = shifts[e * 3 + 0]; b1 = shifts[e * 3 + 1]; }
    else           { b0 = shifts[e * 3 + 2]; b1 = 0.f;               }
    v2f b; b.x = b0; b.y = b1;

    v8f c = {};
    // (neg_a, A, neg_b, B, c_mod, C, reuse_a, reuse_b) -- all 32 lanes active
    c = __builtin_amdgcn_wmma_f32_16x16x4_f32(false, a, false, b,
                                              (short)0, c, false, false);

    // Lanes 0-15 hold dot(k, S_e) for k=0..3 in c[0..3]; exp(-2*pi*i*x).
    if (half == 0) {
        const float TWO_PI = 6.283185307179586476925f;
        #pragma unroll
        for (int k = 0; k < NK; ++k) {
            float ang = -TWO_PI * c[k];
            float s, cc;
            __sincosf(ang, &s, &cc);
            phase[k * NEDGES + e] = make_float2(cc, s);
        }
    }
}

// ---------------------------------------------------------------------------
// 3) Onsite diagonal blocks. After H + conj(H^T):
//    H[k, n*9+r, n*9+c].re = 0.5*(ons[r,c] + ons[c,r]), imag = 0.
//    Each thread uniquely owns (n,r,c); edges never hit diagonal blocks
//    (ej != ei by construction), so plain stores suffice (after zero_kernel).
// ---------------------------------------------------------------------------
__global__ void onsite_kernel(const float* __restrict__ ons,
                              float* __restrict__ H)
{
    int t = blockIdx.x * blockDim.x + threadIdx.x;
    if (t >= NATOMS * 81) return;
    int n  = t / 81, rc = t % 81;
    int r  = rc / NORB, cc = rc % NORB;
    float v = 0.5f * (ons[n * NODED + NODE_IDX[r * NORB + cc]] +
                      ons[n * NODED + NODE_IDX[cc * NORB + r]]);
    int row = n * NORB + r;
    int col = n * NORB + cc;
    #pragma unroll
    for (int k = 0; k < NK; ++k) {
        int idx = ((k * ADIM + row) * ADIM + col) * 2;   // fits in int31
        H[idx] = v;                                      // imag stays 0
    }
}

// ---------------------------------------------------------------------------
// 4) Edge scatter: val = phase[k,e]*hop[e,r,c] at (er,ec), conj(val) at (ec,er).
//    Atomics: duplicate (ei,ej) edges and (a,b)/(b,a) collisions are possible.
// ---------------------------------------------------------------------------
__global__ void edge_kernel(const float*  __restrict__ hop,
                            const int*    __restrict__ eidx,   // [2][NEDGES]
                            const float2* __restrict__ phase,  // [NK][NEDGES]
                            float* __restrict__ H)
{
    int t = blockIdx.x * blockDim.x + threadIdx.x;
    if (t >= NEDGES * 81) return;
    int e  = t / 81, rc = t % 81;
    int r  = rc / NORB, cc = rc % NORB;

    float h  = hop[e * PAIRD + PAIR_IDX[rc]];
    int   ei = eidx[e];
    int   ej = eidx[NEDGES + e];
    int  row = ei * NORB + r;
    int  col = ej * NORB + cc;

    #pragma unroll
    for (int k = 0; k < NK; ++k) {
        float2 p  = phase[k * NEDGES + e];
        float  re = p.x * h;
        float  im = p.y * h;
        int i1 = ((k * ADIM + row) * ADIM + col) * 2;
        int i2 = ((k * ADIM + col) * ADIM + row) * 2;
        atomicAdd(&H[i1],     re);
        atomicAdd(&H[i1 + 1], im);
        atomicAdd(&H[i2],     re);     // conj at mirrored site
        atomicAdd(&H[i2 + 1], -im);
    }
}

// ---------------------------------------------------------------------------
extern "C" void kernel_launch(void* const* d_in, const int* in_sizes, int n_in,
                              void* d_out, int out_size, void* d_ws, size_t ws_size,
                              hipStream_t stream) {
    const float* hop    = (const float*)d_in[0];  // [NEDGES][81]
    const float* ons    = (const float*)d_in[1];  // [NATOMS][58]
    const float* kp     = (const float*)d_in[2];  // [NK][3]
    const int*   eidx   = (const int*)  d_in[3];  // [2][NEDGES]
    const float* shifts = (const float*)d_in[4];  // [NEDGES][3]

    float*  H     = (float*)d_out;                // complex64 interleaved
    float2* phase = (float2*)d_ws;                // NK*NEDGES float2 = 200 KB

    long n4 = (long)out_size / 4;                 // out_size divisible by 4
    zero_kernel<<<2048, 256, 0, stream>>>((float4*)d_out, n4);

    phase_wmma_kernel<<<NEDGES / 16, 32, 0, stream>>>(kp, shifts, phase);

    onsite_kernel<<<(NATOMS * 81 + 255) / 256, 256, 0, stream>>>(ons, H);

    edge_kernel<<<(NEDGES * 81 + 255) / 256, 256, 0, stream>>>(hop, eidx, phase, H);
}